// WBCEWithLogitLoss_3049426780427
// MI455X (gfx1250) — compile-verified
//
#include <hip/hip_runtime.h>
#include <hip/hip_bf16.h>

// -------- CDNA5 async global->LDS path (probe-guarded) ---------------------
#if defined(__has_builtin)
#  if __has_builtin(__builtin_amdgcn_global_load_async_to_lds_b128)
#    define HAVE_ASYNC_LDS 1
#  endif
#  if __has_builtin(__builtin_amdgcn_s_wait_asynccnt)
#    define HAVE_WAIT_ASYNC 1
#  endif
#endif
#ifndef HAVE_ASYNC_LDS
#define HAVE_ASYNC_LDS 0
#endif

#if HAVE_ASYNC_LDS
// Probe round 1 revealed the signature:
//   void __builtin_amdgcn_global_load_async_to_lds_b128(
//       v4i addrspace(1)* src, v4i addrspace(3)* dst, Ii offset, Ii cpol)
typedef int v4i __attribute__((vector_size(16)));
typedef __attribute__((address_space(1))) v4i g_v4i;
typedef __attribute__((address_space(3))) v4i l_v4i;
#define ASYNC_B128(gsrc, ldst)                                                \
  __builtin_amdgcn_global_load_async_to_lds_b128((g_v4i*)(gsrc),              \
                                                 (l_v4i*)(ldst), 0, 0)
#if defined(HAVE_WAIT_ASYNC)
#define WAIT_ASYNC(n)                                                         \
  do {                                                                        \
    asm volatile("" ::: "memory");                                            \
    __builtin_amdgcn_s_wait_asynccnt(n);                                      \
    asm volatile("" ::: "memory");                                            \
  } while (0)
#else
#define WAIT_ASYNC(n) asm volatile("s_wait_asynccnt %0" ::"n"(n) : "memory")
#endif
#endif  // HAVE_ASYNC_LDS

// ---------------------------------------------------------------------------
// result = -(1/B) * sum_j (B-j) * c_j,
// c_j = sum_i [ wp_i*t_i*log(p_i) + wn_i*(1-t_i)*log(1-p_i) ]
// ln2 folded into wpl/wnl so each log is one v_log_f32 (log2).
// ---------------------------------------------------------------------------

__device__ __forceinline__ float sample_nll(const float4 l0, const float4 l1,
                                            const float4 t0, const float4 t1,
                                            const float* __restrict__ wpl,
                                            const float* __restrict__ wnl) {
  const float lp[8] = {l0.x, l0.y, l0.z, l0.w, l1.x, l1.y, l1.z, l1.w};
  const float tt[8] = {t0.x, t0.y, t0.z, t0.w, t1.x, t1.y, t1.z, t1.w};
  float c = 0.0f;
#pragma unroll
  for (int i = 0; i < 8; ++i) {
    const float p = lp[i];
    const float t = tt[i];
    const float a = t * wpl[i];                         // wp*ln2 if t==1 else 0
    const float b = __builtin_fmaf(-t, wnl[i], wnl[i]); // wn*ln2*(1-t), exact
    c = __builtin_fmaf(a, __log2f(p), c);
    c = __builtin_fmaf(b, __log2f(1.0f - p), c);
  }
  return c;
}

#define BLOCK_THREADS 256
#define TILE_SAMPLES 256  // per block per tile (32 per wave)
#define WAVE_SAMPLES 32
#define SAMPLE_BYTES 32    // 8 f32 per sample
#define STAGE_BYTES 2048   // per wave per stage: 1KB logits + 1KB targets

__global__ __launch_bounds__(BLOCK_THREADS) void wbce_partial_kernel(
    const float* __restrict__ logits, const float* __restrict__ targets,
    const float* __restrict__ wp, const float* __restrict__ wn,
    double* __restrict__ partial, float* __restrict__ out, long long B) {
  __shared__ __align__(16) unsigned char smem[8 * 2 * STAGE_BYTES];  // 32 KB
  __shared__ double red[8];

  // Per-class weights, ln2 folded in (uniform -> scalar loads).
  float wpl[8], wnl[8];
#pragma unroll
  for (int i = 0; i < 8; ++i) {
    wpl[i] = wp[i] * 0.69314718056f;
    wnl[i] = wn[i] * 0.69314718056f;
  }

  const int tid = threadIdx.x;
  const int w = tid >> 5;    // wave in block (8 waves)
  const int lane = tid & 31; // wave32
  double acc = 0.0;

  long long tail_start = 0;
#if HAVE_ASYNC_LDS
  {
    const long long T = B / TILE_SAMPLES;  // full tiles
    tail_start = T * (long long)TILE_SAMPLES;
    unsigned char* sb0 = smem + (size_t)(w * 2 + 0) * STAGE_BYTES;
    unsigned char* sb1 = smem + (size_t)(w * 2 + 1) * STAGE_BYTES;
    const int lo = lane * 16;

    auto issue = [&](long long tile, unsigned char* sb) {
      const size_t base =
          ((size_t)tile * TILE_SAMPLES + (size_t)w * WAVE_SAMPLES) *
          SAMPLE_BYTES;
      const unsigned char* gl = (const unsigned char*)logits + base;
      const unsigned char* gt = (const unsigned char*)targets + base;
      // 4 x b128 per lane: 1KB logits + 1KB targets per wave per tile
      ASYNC_B128(gl + lo, sb + lo);
      ASYNC_B128(gl + 512 + lo, sb + 512 + lo);
      ASYNC_B128(gt + lo, sb + 1024 + lo);
      ASYNC_B128(gt + 512 + lo, sb + 1536 + lo);
    };

    long long t = blockIdx.x;
    const long long tstep = gridDim.x;
    if (t < T) issue(t, sb0);  // prologue: stage 0 in flight
    int stage = 0;
    for (; t < T; t += tstep, stage ^= 1) {
      const long long tn = t + tstep;
      unsigned char* sb = stage ? sb1 : sb0;
      unsigned char* sbn = stage ? sb0 : sb1;
      if (tn < T) {
        issue(tn, sbn);  // keep 8 async ops outstanding
        WAIT_ASYNC(4);   // in-order completion => current tile's 4 are done
      } else {
        WAIT_ASYNC(0);
      }
      const float4* l4 = (const float4*)(sb + lane * SAMPLE_BYTES);
      const float4* t4 = (const float4*)(sb + 1024 + lane * SAMPLE_BYTES);
      const float4 L0 = l4[0], L1 = l4[1];
      const float4 T0 = t4[0], T1 = t4[1];
      const float c = sample_nll(L0, L1, T0, T1, wpl, wnl);
      const long long j =
          t * (long long)TILE_SAMPLES + (long long)w * WAVE_SAMPLES + lane;
      acc += (double)(int)(B - j) * (double)c;  // f64 accumulate (exact-ish)
    }
  }
#endif
  // Tail samples (or the whole range if async builtins are unavailable).
  {
    const long long gtid = (long long)blockIdx.x * blockDim.x + tid;
    const long long gstride = (long long)gridDim.x * blockDim.x;
    const float4* lg4 = (const float4*)logits;
    const float4* tg4 = (const float4*)targets;
    for (long long j = tail_start + gtid; j < B; j += gstride) {
      const float4 L0 = lg4[2 * j], L1 = lg4[2 * j + 1];
      const float4 T0 = tg4[2 * j], T1 = tg4[2 * j + 1];
      const float c = sample_nll(L0, L1, T0, T1, wpl, wnl);
      acc += (double)(int)(B - j) * (double)c;
    }
  }

  // wave32 reduction, then cross-wave via LDS
#pragma unroll
  for (int off = 16; off > 0; off >>= 1) acc += __shfl_down(acc, off);
  if (lane == 0) red[w] = acc;
  __syncthreads();
  if (tid == 0) {
    double s = 0.0;
#pragma unroll
    for (int i = 0; i < 8; ++i) s += red[i];
    if (gridDim.x == 1)
      out[0] = (float)(-s / (double)B);
    else
      partial[blockIdx.x] = s;
  }
}

__global__ __launch_bounds__(256) void wbce_finalize_kernel(
    const double* __restrict__ partial, int n, float* __restrict__ out,
    long long B) {
  __shared__ double s[256];
  double a = 0.0;
  for (int i = threadIdx.x; i < n; i += 256) a += partial[i];
  s[threadIdx.x] = a;
  __syncthreads();
  for (int k = 128; k > 0; k >>= 1) {
    if ((int)threadIdx.x < k) s[threadIdx.x] += s[threadIdx.x + k];
    __syncthreads();
  }
  if (threadIdx.x == 0) out[0] = (float)(-s[0] / (double)B);
}

extern "C" void kernel_launch(void* const* d_in, const int* in_sizes, int n_in,
                              void* d_out, int out_size, void* d_ws,
                              size_t ws_size, hipStream_t stream) {
  const float* logits = (const float*)d_in[0];
  const float* targets = (const float*)d_in[1];
  const float* wp = (const float*)d_in[2];
  const float* wn = (const float*)d_in[3];
  float* out = (float*)d_out;
  const long long B = (long long)in_sizes[0] / 8;
  double* partial = (double*)d_ws;

  int nblocks = 2048;  // 8 blocks/WGP (32KB LDS each) on ~256 WGPs
  const size_t cap = ws_size / sizeof(double);
  if ((size_t)nblocks > cap) nblocks = (int)cap;
  if (nblocks < 1) nblocks = 1;

  wbce_partial_kernel<<<nblocks, BLOCK_THREADS, 0, stream>>>(
      logits, targets, wp, wn, partial, out, B);
  if (nblocks > 1)
    wbce_finalize_kernel<<<1, 256, 0, stream>>>(partial, nblocks, out, B);
}